// DepthmapRenderer_38946763440609
// MI455X (gfx1250) — compile-verified
//
#include <hip/hip_runtime.h>
#include <math.h>

#define EPS 1e-8f
#define BIG 1e10f
#define CHUNK_F 64

typedef __attribute__((ext_vector_type(2))) float v2f;
typedef __attribute__((ext_vector_type(8))) float v8f;
typedef __attribute__((ext_vector_type(4))) int   v4i;

#if defined(__has_builtin)
#if __has_builtin(__builtin_amdgcn_global_load_async_to_lds_b128)
#define HAVE_ASYNC_LDS 1
#endif
#endif

#ifdef HAVE_ASYNC_LDS
typedef __attribute__((address_space(1))) v4i* gv4i_p;  // global
typedef __attribute__((address_space(3))) v4i* lv4i_p;  // LDS
#endif

// ---------------------------------------------------------------------------
// Kernel 1: vertex projection via V_WMMA_F32_16X16X4_F32.
// One wave (32 lanes) handles a tile of 16 vertices for one batch.
//   A (16x4) = [x y z 1] per vertex row
//   B (4x16) = B[k][n] = R[n][k] for k<3, t[n] for k==3  (n<3 used)
//   D (16x16): D[m][n] = cam component n of vertex m  (n = 0..2)
// A layout (ISA 16x4 f32 table): lanes 0-15 hold K=0 (VGPR0) / K=1 (VGPR1),
// lanes 16-31 hold K=2 (VGPR0) / K=3 (VGPR1). B mirrored.
// D layout: VGPR r, lane L -> element (M = r + 8*(L>>4), N = L&15).
// ---------------------------------------------------------------------------
__global__ __launch_bounds__(32) void project_kernel(
    const float* __restrict__ mesh, const float* __restrict__ R,
    const float* __restrict__ t, const float* __restrict__ focal,
    const float* __restrict__ princpt,
    float* __restrict__ projx, float* __restrict__ projy,
    float* __restrict__ projz, int B, int V, int tilesPerBatch)
{
    int blk  = blockIdx.x;
    int b    = blk / tilesPerBatch;
    int tile = blk % tilesPerBatch;
    int lane = threadIdx.x;      // 0..31, wave32
    int half = lane >> 4;        // 0: K=0/1   1: K=2/3
    int m16  = lane & 15;

    // ---- A matrix: vertex row [x y z 1]
    int v = tile * 16 + m16;
    float mx = 0.f, my = 0.f, mz = 0.f;
    if (v < V) {
        const float* mv = mesh + ((size_t)b * V + v) * 3;
        mx = mv[0]; my = mv[1]; mz = mv[2];
    }
    v2f A;
    A.x = (half == 0) ? mx : mz;     // K=0 | K=2
    A.y = (half == 0) ? my : 1.0f;   // K=1 | K=3

    // ---- B matrix: columns n<3 carry [R[n][0..2], t[n]]
    int n = m16;
    float b0 = 0.f, b1 = 0.f;
    if (n < 3) {
        const float* Rb = R + b * 9;
        b0 = (half == 0) ? Rb[n * 3 + 0] : Rb[n * 3 + 2];  // K=0 | K=2
        b1 = (half == 0) ? Rb[n * 3 + 1] : t[b * 3 + n];   // K=1 | K=3
    }
    v2f Bm; Bm.x = b0; Bm.y = b1;

    v8f C = {};
    v8f D = __builtin_amdgcn_wmma_f32_16x16x4_f32(
        /*neg_a=*/false, A, /*neg_b=*/false, Bm,
        /*c_mod=*/(short)0, C, /*reuse_a=*/false, /*reuse_b=*/false);

    // Scatter D -> per-vertex cam[xyz] through LDS (columns 0..2 only).
    __shared__ float cam[16][4];
    if (n < 3) {
#pragma unroll
        for (int r = 0; r < 8; ++r)
            cam[r + 8 * half][n] = D[r];
    }
    __syncthreads();

    if (half == 0 && v < V) {
        float cx = cam[m16][0], cy = cam[m16][1], cz = cam[m16][2];
        float zs = (fabsf(cz) > EPS) ? cz : EPS;
        float px = focal[b * 2 + 0] * cx / zs + princpt[b * 2 + 0];
        float py = focal[b * 2 + 1] * cy / zs + princpt[b * 2 + 1];
        size_t o = (size_t)b * V + v;
        projx[o] = px; projy[o] = py; projz[o] = cz;
    }
}

// ---------------------------------------------------------------------------
// Kernel 2: per-face edge-function coefficients (hoists all per-face algebra
// out of the 65536-pixel loop). Record = 16 floats:
//  [A0 B0 C0  A1 B1 C1  A2 B2 C2  Az Bz Cz  inv_area  valid  0 0]
//  wk(px,py) = Ak*px + Bk*py + Ck ;  zi = (Az*px+Bz*py+Cz)*inv_area
// ---------------------------------------------------------------------------
__global__ __launch_bounds__(256) void face_setup_kernel(
    const float* __restrict__ projx, const float* __restrict__ projy,
    const float* __restrict__ projz, const int* __restrict__ face,
    float* __restrict__ rec, int B, int V, int F)
{
    int idx = blockIdx.x * blockDim.x + threadIdx.x;
    if (idx >= B * F) return;
    int b = idx / F, f = idx % F;
    int i0 = face[f * 3 + 0], i1 = face[f * 3 + 1], i2 = face[f * 3 + 2];
    size_t base = (size_t)b * V;
    float x0 = projx[base + i0], y0 = projy[base + i0], z0 = projz[base + i0];
    float x1 = projx[base + i1], y1 = projy[base + i1], z1 = projz[base + i1];
    float x2 = projx[base + i2], y2 = projy[base + i2], z2 = projz[base + i2];

    // w0 = (x2-x1)*(py-y1) - (y2-y1)*(px-x1)
    float e0x = x2 - x1, e0y = y2 - y1;
    float A0 = -e0y, B0 = e0x, C0 = e0y * x1 - e0x * y1;
    // w1 = (x0-x2)*(py-y2) - (y0-y2)*(px-x2)
    float e1x = x0 - x2, e1y = y0 - y2;
    float A1 = -e1y, B1 = e1x, C1 = e1y * x2 - e1x * y2;
    // w2 = (x1-x0)*(py-y0) - (y1-y0)*(px-x0)
    float e2x = x1 - x0, e2y = y1 - y0;
    float A2 = -e2y, B2 = e2x, C2 = e2y * x0 - e2x * y0;

    float area   = e2x * (y2 - y0) - e2y * (x2 - x0);
    float area_s = (fabsf(area) > EPS) ? area : 1.0f;
    float Az = A0 * z0 + A1 * z1 + A2 * z2;
    float Bz = B0 * z0 + B1 * z1 + B2 * z2;
    float Cz = C0 * z0 + C1 * z1 + C2 * z2;
    float zmin  = fminf(z0, fminf(z1, z2));
    float valid = ((zmin > EPS) && (fabsf(area) > EPS)) ? 1.0f : 0.0f;

    float* r = rec + (size_t)idx * 16;
    r[0] = A0;  r[1] = B0;  r[2]  = C0;
    r[3] = A1;  r[4] = B1;  r[5]  = C1;
    r[6] = A2;  r[7] = B2;  r[8]  = C2;
    r[9] = Az;  r[10] = Bz; r[11] = Cz;
    r[12] = 1.0f / area_s;  r[13] = valid;
    r[14] = 0.f; r[15] = 0.f;
}

// ---------------------------------------------------------------------------
// Kernel 3: rasterize. 256 threads = 256 pixels per block (one image row).
// Faces staged to LDS in 64-face / 4KB chunks via CDNA5 async-to-LDS
// (ASYNCcnt path), then a pure-FMA inner loop; z-buffer lives in a register.
// ---------------------------------------------------------------------------
__global__ __launch_bounds__(256) void raster_kernel(
    const float* __restrict__ rec, float* __restrict__ out,
    int B, int F, int H, int W, int blocksPerBatch)
{
    __shared__ float lds[CHUNK_F * 16];
    int b    = blockIdx.x / blocksPerBatch;
    int tile = blockIdx.x % blocksPerBatch;
    int HW   = H * W;
    int pix  = tile * 256 + threadIdx.x;
    bool live = (pix < HW);
    int row = live ? (pix / W) : 0;
    int col = live ? (pix % W) : 0;
    float px = (float)col + 0.5f;
    float py = (float)row + 0.5f;
    float zbuf = BIG;
    const float* recB = rec + (size_t)b * F * 16;

    for (int f0 = 0; f0 < F; f0 += CHUNK_F) {
        int nf = F - f0; if (nf > CHUNK_F) nf = CHUNK_F;
        int nq = nf * 4;  // 16-byte quads in this chunk
        if ((int)threadIdx.x < nq) {
            const float* gsrc = recB + (size_t)f0 * 16 + (size_t)threadIdx.x * 4;
            float*       ldst = lds + threadIdx.x * 4;
#ifdef HAVE_ASYNC_LDS
            __builtin_amdgcn_global_load_async_to_lds_b128(
                (gv4i_p)gsrc, (lv4i_p)ldst, 0, 0);
#else
            *(float4*)ldst = *(const float4*)gsrc;
#endif
        }
#ifdef HAVE_ASYNC_LDS
#if __has_builtin(__builtin_amdgcn_s_wait_asynccnt)
        __builtin_amdgcn_s_wait_asynccnt(0);
#else
        asm volatile("s_wait_asynccnt 0" ::: "memory");
#endif
#endif
        __syncthreads();

        for (int j = 0; j < nf; ++j) {
            const float* c = lds + j * 16;
            float w0 = fmaf(c[0], px, fmaf(c[1],  py, c[2]));
            float w1 = fmaf(c[3], px, fmaf(c[4],  py, c[5]));
            float w2 = fmaf(c[6], px, fmaf(c[7],  py, c[8]));
            float zi = fmaf(c[9], px, fmaf(c[10], py, c[11])) * c[12];
            bool inside = (w0 >= 0.f && w1 >= 0.f && w2 >= 0.f) ||
                          (w0 <= 0.f && w1 <= 0.f && w2 <= 0.f);
            bool ok = inside && (c[13] > 0.5f) && (zi > EPS);
            zbuf = ok ? fminf(zbuf, zi) : zbuf;
        }
        __syncthreads();
    }

    if (live)
        out[(size_t)b * HW + pix] = (zbuf < 0.5f * BIG) ? zbuf : -1.0f;
}

// ---------------------------------------------------------------------------
extern "C" void kernel_launch(void* const* d_in, const int* in_sizes, int n_in,
                              void* d_out, int out_size, void* d_ws, size_t ws_size,
                              hipStream_t stream)
{
    const float* mesh    = (const float*)d_in[0];
    const float* R       = (const float*)d_in[1];
    const float* t       = (const float*)d_in[2];
    const float* focal   = (const float*)d_in[3];
    const float* princpt = (const float*)d_in[4];
    const int*   face    = (const int*)d_in[5];

    int B = in_sizes[1] / 9;            // R is (B,3,3)
    int V = in_sizes[0] / (3 * B);      // mesh is (B,V,3)
    int F = in_sizes[5] / 3;            // face is (F,3)
    int HW = out_size / B;              // out is (B,1,H,W)
    int W = 1; while (W * W < HW) ++W;  // square image
    int H = HW / W;

    // Workspace layout (floats): [projx | projy | projz | face records]
    float* ws    = (float*)d_ws;
    float* projx = ws;
    float* projy = projx + (size_t)B * V;
    float* projz = projy + (size_t)B * V;
    size_t projTot = (size_t)3 * B * V;
    size_t recOff  = (projTot + 63) & ~(size_t)63;
    float* rec     = ws + recOff;

    int tilesPB = (V + 15) / 16;
    hipLaunchKernelGGL(project_kernel, dim3(B * tilesPB), dim3(32), 0, stream,
                       mesh, R, t, focal, princpt, projx, projy, projz,
                       B, V, tilesPB);

    int nBF = B * F;
    hipLaunchKernelGGL(face_setup_kernel, dim3((nBF + 255) / 256), dim3(256), 0,
                       stream, projx, projy, projz, face, rec, B, V, F);

    int blocksPB = (HW + 255) / 256;
    hipLaunchKernelGGL(raster_kernel, dim3(B * blocksPB), dim3(256), 0, stream,
                       rec, (float*)d_out, B, F, H, W, blocksPB);
}